// IntraClipMerging_12266426598093
// MI455X (gfx1250) — compile-verified
//
#include <hip/hip_runtime.h>

#define B_    32
#define N_    1024
#define D_    512
#define K_SEL 512
#define EPS_  1e-8f

typedef __attribute__((ext_vector_type(2))) float v2f;
typedef __attribute__((ext_vector_type(8))) float v8f;

// ---------------------------------------------------------------------------
// k1: per-token inverse norms + per-block partial sums of unit vectors.
// One block = 64 rows of one batch. 8 waves, 8 rows per wave, 16 dims/lane.
// ---------------------------------------------------------------------------
__global__ __launch_bounds__(256)
void k1_norms_part(const float* __restrict__ x, float* __restrict__ inv_norms,
                   float* __restrict__ part) {
  const int b      = blockIdx.x >> 4;          // 16 row-blocks per batch
  const int rowblk = blockIdx.x & 15;
  const int row0   = rowblk << 6;              // *64
  const int lane   = threadIdx.x & 31;
  const int wave   = threadIdx.x >> 5;         // 0..7

  __shared__ float lds_acc[8][512];

  float acc[16];
#pragma unroll
  for (int i = 0; i < 16; ++i) acc[i] = 0.f;

  const float* xb = x + (size_t)b * (N_ * D_);

  for (int r = 0; r < 8; ++r) {
    const int row = row0 + wave * 8 + r;
    const float4* xr = (const float4*)(xb + (size_t)row * D_);
    float4 v[4];
    float ss = 0.f;
#pragma unroll
    for (int c = 0; c < 4; ++c) {
      v[c] = xr[c * 32 + lane];                 // d = c*128 + lane*4 .. +3
      ss += v[c].x * v[c].x + v[c].y * v[c].y + v[c].z * v[c].z + v[c].w * v[c].w;
    }
    if (r + 1 < 8)                              // gfx1250 global_prefetch_b8
      __builtin_prefetch(xb + (size_t)(row + 1) * D_ + lane * 16, 0, 1);
#pragma unroll
    for (int off = 16; off > 0; off >>= 1) ss += __shfl_xor(ss, off, 32);
    const float inv = 1.f / fmaxf(sqrtf(ss), EPS_);
    if (lane == 0) inv_norms[b * N_ + row] = inv;
#pragma unroll
    for (int c = 0; c < 4; ++c) {
      acc[c * 4 + 0] += v[c].x * inv;
      acc[c * 4 + 1] += v[c].y * inv;
      acc[c * 4 + 2] += v[c].z * inv;
      acc[c * 4 + 3] += v[c].w * inv;
    }
  }

#pragma unroll
  for (int c = 0; c < 4; ++c) {
    const int d = c * 128 + lane * 4;
    lds_acc[wave][d + 0] = acc[c * 4 + 0];
    lds_acc[wave][d + 1] = acc[c * 4 + 1];
    lds_acc[wave][d + 2] = acc[c * 4 + 2];
    lds_acc[wave][d + 3] = acc[c * 4 + 3];
  }
  __syncthreads();

  for (int d = threadIdx.x; d < 512; d += 256) {
    float s = 0.f;
#pragma unroll
    for (int w8 = 0; w8 < 8; ++w8) s += lds_acc[w8][d];
    part[((size_t)b * 16 + rowblk) * D_ + d] = s;   // deterministic (no atomics)
  }
}

// ---------------------------------------------------------------------------
// k1b: deterministic reduce of the 16 partials -> mean_unit (1/N folded in).
// ---------------------------------------------------------------------------
__global__ __launch_bounds__(256)
void k1b_mu(const float* __restrict__ part, float* __restrict__ mu) {
  const int i = blockIdx.x * 256 + threadIdx.x;     // 0 .. B*D-1
  if (i >= B_ * D_) return;
  const int b = i / D_;
  const int d = i - b * D_;
  float s = 0.f;
#pragma unroll
  for (int j = 0; j < 16; ++j) s += part[((size_t)b * 16 + j) * D_ + d];
  mu[i] = s * (1.0f / N_);
}

// ---------------------------------------------------------------------------
// k2: scores[b,m] = (x[b,m] . mean_unit[b]) * inv_norm[b,m].  L2-resident.
// One wave per row; mean_unit staged in LDS.
// ---------------------------------------------------------------------------
__global__ __launch_bounds__(256)
void k2_scores(const float* __restrict__ x, const float* __restrict__ inv_norms,
               const float* __restrict__ mu, float* __restrict__ scores) {
  const int b    = blockIdx.x >> 7;               // 128 blocks per batch
  const int row0 = (blockIdx.x & 127) << 3;       // 8 rows per block
  const int lane = threadIdx.x & 31;
  const int wave = threadIdx.x >> 5;

  __shared__ float4 mul[128];
  {
    const float4* mub = (const float4*)(mu + (size_t)b * D_);
    for (int i = threadIdx.x; i < 128; i += 256) mul[i] = mub[i];
  }
  __syncthreads();

  const int row = row0 + wave;
  const float4* xr = (const float4*)(x + ((size_t)b * N_ + row) * D_);
  float dot = 0.f;
#pragma unroll
  for (int c = 0; c < 4; ++c) {
    const float4 v = xr[c * 32 + lane];
    const float4 m = mul[c * 32 + lane];
    dot += v.x * m.x + v.y * m.y + v.z * m.z + v.w * m.w;
  }
#pragma unroll
  for (int off = 16; off > 0; off >>= 1) dot += __shfl_xor(dot, off, 32);
  if (lane == 0) scores[b * N_ + row] = dot * inv_norms[b * N_ + row];
}

// ---------------------------------------------------------------------------
// k3: exact top-k membership via rank counting (stable tie-break = lax.top_k).
// w[b,m] = (rank < k) ? 1/k : 0
// ---------------------------------------------------------------------------
__global__ __launch_bounds__(256)
void k3_rank(const float* __restrict__ scores, float* __restrict__ w) {
  const int b = blockIdx.x >> 2;                  // 4 blocks per batch
  const int m = ((blockIdx.x & 3) << 8) + threadIdx.x;

  __shared__ float s[N_];
  for (int i = threadIdx.x; i < N_; i += 256) s[i] = scores[b * N_ + i];
  __syncthreads();

  const float sm = s[m];
  int rank = 0;
  for (int n = 0; n < N_; ++n) {
    const float sn = s[n];
    rank += (sn > sm) || ((sn == sm) && (n < m));
  }
  w[b * N_ + m] = (rank < K_SEL) ? (1.0f / K_SEL) : 0.0f;
}

// ---------------------------------------------------------------------------
// k4: merged[b,d] = sum_m w[m] * x[b,m,d]  as a WMMA chain.
//   A (16x4 f32): weight chunk replicated over M  -> lanes 0-15 hold {w[k],w[k+1]},
//                 lanes 16-31 hold {w[k+2],w[k+3]}   (ISA 16x4 f32 A layout)
//   B (4x16 f32): rows of x striped across lane halves (coalesced 64B segments)
//   C/D chained accumulator; row M=0 (VGPR0, lanes 0-15) is the result.
// One wave owns one 16-dim output tile; K loop = 256 v_wmma_f32_16x16x4_f32.
// ---------------------------------------------------------------------------
__global__ __launch_bounds__(256)
void k4_merge_wmma(const float* __restrict__ x, const float* __restrict__ w,
                   float* __restrict__ out) {
  const int b     = blockIdx.x >> 2;              // 4 blocks per batch
  const int tile  = ((blockIdx.x & 3) << 3) + (threadIdx.x >> 5); // 0..31
  const int lane  = threadIdx.x & 31;
  const int half  = lane >> 4;                    // 0: K=0/1, 1: K=2/3
  const int ln    = lane & 15;
  const int dbase = tile << 4;

  __shared__ float wl[N_];
  for (int i = threadIdx.x; i < N_; i += 256) wl[i] = w[b * N_ + i];
  __syncthreads();

  const float* xb = x + (size_t)b * (N_ * D_);

  v8f c = {};                                      // zero accumulator
  for (int k = 0; k < N_; k += 4) {
    const int ka = k + (half ? 2 : 0);
    const int kb = ka + 1;
    v2f a  = { wl[ka], wl[kb] };                          // A: weights (bcast over M)
    v2f bv = { xb[(size_t)ka * D_ + dbase + ln],          // B row ka, cols dbase..+15
               xb[(size_t)kb * D_ + dbase + ln] };        // B row kb
    c = __builtin_amdgcn_wmma_f32_16x16x4_f32(
        /*neg_a=*/false, a, /*neg_b=*/false, bv,
        /*c_mod=*/(short)0, c, /*reuse_a=*/false, /*reuse_b=*/false);
  }

  if (lane < 16) out[(size_t)b * D_ + dbase + ln] = c[0];  // M=0 row
}

// ---------------------------------------------------------------------------
extern "C" void kernel_launch(void* const* d_in, const int* in_sizes, int n_in,
                              void* d_out, int out_size, void* d_ws, size_t ws_size,
                              hipStream_t stream) {
  (void)in_sizes; (void)n_in; (void)out_size; (void)ws_size;

  const float* x = (const float*)d_in[0];
  float* out = (float*)d_out;                      // [32, 1, 512] f32

  char* ws = (char*)d_ws;                          // byte offsets (1.44 MB total)
  float* inv_norms = (float*)(ws + 0);             // B*N   = 131072 B
  float* scores    = (float*)(ws + 131072);        // B*N   = 131072 B
  float* wsel      = (float*)(ws + 262144);        // B*N   = 131072 B
  float* mu        = (float*)(ws + 393216);        // B*D   =  65536 B
  float* part      = (float*)(ws + 458752);        // B*16*D = 1 MiB

  k1_norms_part<<<B_ * 16, 256, 0, stream>>>(x, inv_norms, part);
  k1b_mu<<<(B_ * D_ + 255) / 256, 256, 0, stream>>>(part, mu);
  k2_scores<<<B_ * 128, 256, 0, stream>>>(x, inv_norms, mu, scores);
  k3_rank<<<B_ * 4, 256, 0, stream>>>(scores, wsel);
  k4_merge_wmma<<<B_ * 4, 256, 0, stream>>>(x, wsel, out);
}